// SparseResidualBlock_11922829213747
// MI455X (gfx1250) — compile-verified
//
#include <hip/hip_runtime.h>
#include <hip/hip_bf16.h>

typedef float v2f __attribute__((ext_vector_type(2)));
typedef float v8f __attribute__((ext_vector_type(8)));

#define DIM   4096
#define ROOT  64
#define NB    8192

__device__ __forceinline__ v8f wmma4(v2f a, v2f b, v8f c) {
    // D = A(16x4 f32) * B(4x16 f32) + C(16x16 f32)
    return __builtin_amdgcn_wmma_f32_16x16x4_f32(
        /*neg_a=*/false, a, /*neg_b=*/false, b,
        /*c_mod=*/(short)0, c, /*reuse_a=*/false, /*reuse_b=*/false);
}

// --- CDNA5 async global->LDS copy helpers -----------------------------------
typedef const __attribute__((address_space(3))) void* lds_cvptr;

__device__ __forceinline__ unsigned lds_off(const void* p) {
    // generic -> LDS addrspace cast; AS(3) pointers are the raw LDS byte offset
    return (unsigned)(unsigned long long)(lds_cvptr)p;
}

__device__ __forceinline__ void async_copy16(const float* g, const float* l) {
    unsigned loff = lds_off(l);
    // per-lane 16B copy, tracked by ASYNCcnt, bypasses VGPRs
    asm volatile("global_load_async_to_lds_b128 %0, %1, off"
                 :: "v"(loff), "v"(g) : "memory");
}

__device__ __forceinline__ void wait_async0() {
    asm volatile("s_wait_asynccnt 0x0" ::: "memory");
}

// ---------------------------------------------------------------------------
// Kernel 1: block-diagonal MLP (2 layers, relu) + LayerNorm, fused.
// Block = 16 batch rows. 8 waves, each wave handles 8 of the 64 r-blocks.
// Writes normalized activations [NB, DIM] to hout (workspace).
// ---------------------------------------------------------------------------
__global__ __launch_bounds__(256) void k_blockmlp_ln(
    const float* __restrict__ x,
    const float* __restrict__ p1,   // [64,256,64]
    const float* __restrict__ b1,   // [64,256]
    const float* __restrict__ p2,   // [64,64,256]
    const float* __restrict__ b2,   // [64,64]
    const float* __restrict__ gamma,
    const float* __restrict__ beta,
    float* __restrict__ hout)
{
    __shared__ float lds_h1[8][16 * 66];   // wave-private H1 chunk, stride 66
    __shared__ float s_sum[16], s_sq[16], s_mean[16], s_inv[16];

    const int tid  = threadIdx.x;
    const int wid  = tid >> 5;
    const int lane = tid & 31;
    const int lmod = lane & 15;
    const int lhi  = lane >> 4;          // 0 or 1
    const int m0   = blockIdx.x * 16;    // batch-row tile

    if (tid < 16) { s_sum[tid] = 0.0f; s_sq[tid] = 0.0f; }
    __syncthreads();

    float rs[8], rq[8];
#pragma unroll
    for (int v = 0; v < 8; ++v) { rs[v] = 0.0f; rq[v] = 0.0f; }

    float* hl = &lds_h1[wid][0];
    const v8f vzero = {0.f,0.f,0.f,0.f,0.f,0.f,0.f,0.f};

    for (int rr = 0; rr < 8; ++rr) {
        const int r = wid * 8 + rr;

        // Preload X A-fragments for all 16 k-steps (K=64): 16x4 f32 layout.
        v2f xf[16];
        const float* xp = x + (size_t)(m0 + lmod) * DIM + r * ROOT + lhi * 2;
#pragma unroll
        for (int kk = 0; kk < 16; ++kk)
            xf[kk] = *(const v2f*)(xp + kk * 4);

        v8f h2[4];
#pragma unroll
        for (int nt = 0; nt < 4; ++nt) h2[nt] = vzero;

        for (int c = 0; c < 4; ++c) {           // 4 chunks of 64 of the 256 mids
            // ---- layer 1: H1c[16,64] = relu(Xr @ P1_r[chunk]^T + b1) ----
            v8f a1[4];
#pragma unroll
            for (int nt = 0; nt < 4; ++nt) a1[nt] = vzero;

            const float* p1base =
                p1 + ((size_t)r * 256 + c * 64 + lmod) * 64 + lhi * 2;
#pragma unroll
            for (int kk = 0; kk < 16; ++kk) {
#pragma unroll
                for (int nt = 0; nt < 4; ++nt) {
                    v2f bf = *(const v2f*)(p1base + nt * (16 * 64) + kk * 4);
                    a1[nt] = wmma4(xf[kk], bf, a1[nt]);
                }
            }
            // bias + relu, stage into wave-private LDS (A-layout for layer 2)
#pragma unroll
            for (int nt = 0; nt < 4; ++nt) {
                float bb = b1[(size_t)r * 256 + c * 64 + nt * 16 + lmod];
                const int colb = nt * 16 + lmod;
#pragma unroll
                for (int v = 0; v < 8; ++v) {
                    float val = a1[nt][v] + bb;
                    val = val > 0.0f ? val : 0.0f;
                    hl[(v + lhi * 8) * 66 + colb] = val;
                }
            }
            // ---- layer 2 partial: H2 += H1c @ P2_r[:, chunk]^T ----
            const float* p2base =
                p2 + ((size_t)r * 64 + lmod) * 256 + c * 64 + lhi * 2;
#pragma unroll
            for (int kk = 0; kk < 16; ++kk) {
                v2f a2 = *(const v2f*)(hl + lmod * 66 + kk * 4 + lhi * 2);
#pragma unroll
                for (int nt = 0; nt < 4; ++nt) {
                    v2f bf2 = *(const v2f*)(p2base + nt * (16 * 256) + kk * 4);
                    h2[nt] = wmma4(a2, bf2, h2[nt]);
                }
            }
        }

        // bias2 + relu, write pre-LN h, accumulate LN partial sums
#pragma unroll
        for (int nt = 0; nt < 4; ++nt) {
            float bb = b2[r * 64 + nt * 16 + lmod];
            const int col = r * ROOT + nt * 16 + lmod;
#pragma unroll
            for (int v = 0; v < 8; ++v) {
                float val = h2[nt][v] + bb;
                val = fmaxf(val, 0.0f);
                const int row = v + lhi * 8;
                hout[(size_t)(m0 + row) * DIM + col] = val;
                rs[v] += val;
                rq[v] += val * val;
            }
        }
    }

    // cross-wave row-sum reduction in LDS
#pragma unroll
    for (int v = 0; v < 8; ++v) {
        const int row = v + lhi * 8;
        atomicAdd(&s_sum[row], rs[v]);
        atomicAdd(&s_sq[row], rq[v]);
    }
    __threadfence_block();
    __syncthreads();

    if (tid < 16) {
        float mu  = s_sum[tid] * (1.0f / DIM);
        float var = s_sq[tid] * (1.0f / DIM) - mu * mu;
        s_mean[tid] = mu;
        s_inv[tid]  = rsqrtf(var + 1e-5f);
    }
    __syncthreads();

    // normalize the 16xDIM tile in place (L2-hot)
    for (int j = tid; j < 16 * (DIM / 4); j += 256) {
        const int row = j >> 10;                 // DIM/4 = 1024 float4 per row
        const int c4  = (j & 1023) * 4;
        float* hp = &hout[(size_t)(m0 + row) * DIM + c4];
        float4 hv = *(float4*)hp;
        const float mu = s_mean[row], iv = s_inv[row];
        float4 g  = *(const float4*)&gamma[c4];
        float4 bt = *(const float4*)&beta[c4];
        hv.x = (hv.x - mu) * iv * g.x + bt.x;
        hv.y = (hv.y - mu) * iv * g.y + bt.y;
        hv.z = (hv.z - mu) * iv * g.z + bt.z;
        hv.w = (hv.w - mu) * iv * g.w + bt.w;
        *(float4*)hp = hv;
    }
}

// ---------------------------------------------------------------------------
// Kernel 2: out = relu(h_ln @ W^T + b) + x.  128x128 C-tile, KT=16,
// double-buffered LDS via async global->LDS copies (ASYNCcnt),
// 4x2 waves, 32x64 per wave.
// ---------------------------------------------------------------------------
__global__ __launch_bounds__(256) void k_gemm_out(
    const float* __restrict__ A,      // h_ln [NB, DIM]
    const float* __restrict__ W,      // [DIM, DIM], row n is contraction row
    const float* __restrict__ bias,   // [DIM]
    const float* __restrict__ x,      // residual [NB, DIM]
    float* __restrict__ out)
{
    __shared__ float sA[2][128 * 20];
    __shared__ float sB[2][128 * 20];

    const int tid  = threadIdx.x;
    const int wid  = tid >> 5;
    const int lane = tid & 31;
    const int lmod = lane & 15;
    const int lhi  = lane >> 4;
    const int nB0  = blockIdx.x * 128;
    const int mB0  = blockIdx.y * 128;
    const int wm   = wid >> 1;          // 0..3
    const int wn   = wid & 1;           // 0..1

    v8f acc[2][4];
    const v8f vzero = {0.f,0.f,0.f,0.f,0.f,0.f,0.f,0.f};
#pragma unroll
    for (int mf = 0; mf < 2; ++mf)
#pragma unroll
        for (int nf = 0; nf < 4; ++nf) acc[mf][nf] = vzero;

    // tile fill: 512 float4 per tile per matrix; 2 per thread per matrix
    const int f0row = tid >> 2;            // flat idx tid
    const int f1row = (tid + 256) >> 2;    // flat idx tid+256
    const int f0c   = (tid & 3) * 4;
    const int f1c   = ((tid + 256) & 3) * 4;

    const float* Arow0 = &A[(size_t)(mB0 + f0row) * DIM + f0c];
    const float* Arow1 = &A[(size_t)(mB0 + f1row) * DIM + f1c];
    const float* Wrow0 = &W[(size_t)(nB0 + f0row) * DIM + f0c];
    const float* Wrow1 = &W[(size_t)(nB0 + f1row) * DIM + f1c];

    auto issue_tile = [&](int kt, int p) {
        const int kb = kt * 16;
        async_copy16(Arow0 + kb, &sA[p][f0row * 20 + f0c]);
        async_copy16(Arow1 + kb, &sA[p][f1row * 20 + f1c]);
        async_copy16(Wrow0 + kb, &sB[p][f0row * 20 + f0c]);
        async_copy16(Wrow1 + kb, &sB[p][f1row * 20 + f1c]);
    };

    issue_tile(0, 0);
    wait_async0();
    __syncthreads();

    const int KT = DIM / 16;              // 256
    for (int kt = 0; kt < KT; ++kt) {
        const int p = kt & 1;
        if (kt + 1 < KT) issue_tile(kt + 1, p ^ 1);
        if (kt + 8 < KT) {                // pre-stage L2 ahead of the async fills
            __builtin_prefetch(Arow0 + (kt + 8) * 16, 0, 0);
            __builtin_prefetch(Wrow0 + (kt + 8) * 16, 0, 0);
        }

#pragma unroll
        for (int ks = 0; ks < 16; ks += 4) {
            v2f af[2], bf[4];
#pragma unroll
            for (int mf = 0; mf < 2; ++mf)
                af[mf] = *(const v2f*)
                    &sA[p][(wm * 32 + mf * 16 + lmod) * 20 + ks + lhi * 2];
#pragma unroll
            for (int nf = 0; nf < 4; ++nf)
                bf[nf] = *(const v2f*)
                    &sB[p][(wn * 64 + nf * 16 + lmod) * 20 + ks + lhi * 2];
#pragma unroll
            for (int mf = 0; mf < 2; ++mf)
#pragma unroll
                for (int nf = 0; nf < 4; ++nf)
                    acc[mf][nf] = wmma4(af[mf], bf[nf], acc[mf][nf]);
        }

        if (kt + 1 < KT) wait_async0();   // next buffer filled before swap
        __syncthreads();
    }

    // epilogue: + bias, relu, + residual
#pragma unroll
    for (int nf = 0; nf < 4; ++nf) {
        const int col = nB0 + wn * 64 + nf * 16 + lmod;
        const float bb = bias[col];
#pragma unroll
        for (int mf = 0; mf < 2; ++mf) {
#pragma unroll
            for (int v = 0; v < 8; ++v) {
                const int row = mB0 + wm * 32 + mf * 16 + v + lhi * 8;
                float val = acc[mf][nf][v] + bb;
                val = fmaxf(val, 0.0f);
                val += x[(size_t)row * DIM + col];
                out[(size_t)row * DIM + col] = val;
            }
        }
    }
}

extern "C" void kernel_launch(void* const* d_in, const int* in_sizes, int n_in,
                              void* d_out, int out_size, void* d_ws, size_t ws_size,
                              hipStream_t stream) {
    const float* x     = (const float*)d_in[0];
    const float* p1    = (const float*)d_in[1];
    const float* b1    = (const float*)d_in[2];
    const float* p2    = (const float*)d_in[3];
    const float* b2    = (const float*)d_in[4];
    const float* gamma = (const float*)d_in[5];
    const float* beta  = (const float*)d_in[6];
    const float* W     = (const float*)d_in[7];
    const float* b     = (const float*)d_in[8];
    float* out = (float*)d_out;
    float* hln = (float*)d_ws;            // NB*DIM fp32 = 128 MB

    k_blockmlp_ln<<<NB / 16, 256, 0, stream>>>(x, p1, b1, p2, b2, gamma, beta, hln);

    dim3 grid(DIM / 128, NB / 128);       // (32, 64)
    k_gemm_out<<<grid, 256, 0, stream>>>(hln, W, b, x, out);
}